// Router_25941602468241
// MI455X (gfx1250) — compile-verified
//
#include <hip/hip_runtime.h>
#include <hip/hip_bf16.h>

// MoE per-token expert apply: out[t] = x[t] @ W[split[t]] + b[split[t]]
// N=262144 tokens, D_IN=D_OUT=128, E=8 experts, all fp32.
// Bucket tokens by expert into 16-aligned segments, then fp32 WMMA GEMM.

typedef __attribute__((ext_vector_type(2))) float v2f;
typedef __attribute__((ext_vector_type(8))) float v8f;

#define N_EXPERTS 8
#define D_IN      128
#define D_OUT     128
#define TILE_M    16
#define BPE       64          // blocks per expert in GEMM kernel
#define GEMM_BLK  256         // 8 waves of 32
// W in LDS is stored K-paired: w_pair[p][n][2] holds (W[2p][n], W[2p+1][n]).
// kpair stride 288 floats: 288%64==32 -> the two half-waves (kpair p vs p+1)
// land in disjoint bank ranges; within a half-wave each lane owns its own
// even/odd bank pair. One aligned ds_load_b64 per B operand, zero shuffles.
#define WP_STRIDE 288
#define A_LDS_STRIDE 132      // 128 + 4: conflict-free b64 column reads

// ws (int32) layout:
//  [0..7]   counts
//  [8..15]  segStart (token units, 16-aligned)
//  [16..23] tileCount per expert
//  [24..31] scatter cursors
//  [32...]  perm array (N + 8*16 entries, pads = -1)
#define WS_COUNT  0
#define WS_SEG    8
#define WS_TILES  16
#define WS_CUR    24
#define WS_PERM   32

__global__ void moe_init_kernel(int* ws) {
    int t = threadIdx.x;
    if (t < 32) ws[t] = 0;
}

__global__ void moe_hist_kernel(const long long* __restrict__ split, int* ws, int n) {
    __shared__ int hc[N_EXPERTS];
    int tid = threadIdx.x;
    if (tid < N_EXPERTS) hc[tid] = 0;
    __syncthreads();
    int t = blockIdx.x * blockDim.x + tid;
    if (t < n) {
        int e = (int)split[t];
        atomicAdd(&hc[e], 1);
    }
    __syncthreads();
    if (tid < N_EXPERTS) atomicAdd(&ws[WS_COUNT + tid], hc[tid]);
}

__global__ void moe_scan_pad_kernel(int* ws) {
    if (threadIdx.x != 0) return;
    int* perm = ws + WS_PERM;
    int running = 0;
    for (int e = 0; e < N_EXPERTS; ++e) {
        int c  = ws[WS_COUNT + e];
        int Te = (c + TILE_M - 1) / TILE_M;
        ws[WS_SEG + e]   = running;
        ws[WS_TILES + e] = Te;
        for (int p = c; p < Te * TILE_M; ++p) perm[running + p] = -1;
        running += Te * TILE_M;
    }
}

__global__ void moe_scatter_kernel(const long long* __restrict__ split, int* ws, int n) {
    int t = blockIdx.x * blockDim.x + threadIdx.x;
    if (t >= n) return;
    int e = (int)split[t];
    int pos = ws[WS_SEG + e] + atomicAdd(&ws[WS_CUR + e], 1);
    ws[WS_PERM + pos] = t;
}

// GEMM: block = (expert e, slice j). Stage W[e] K-paired in LDS once, then 8
// waves independently process 16-token tiles of expert e's segment.
__global__ void __launch_bounds__(GEMM_BLK)
moe_gemm_kernel(const float* __restrict__ x,
                const float* __restrict__ W,
                const float* __restrict__ bias,
                const int*   __restrict__ ws,
                float* __restrict__ out) {
    extern __shared__ float smem[];
    float* w_pair = smem;                              // 64 kpairs x 288 floats
    float* a_lds  = smem + (D_IN / 2) * WP_STRIDE;     // 8 waves x 16 x 132 floats

    const int e   = blockIdx.x / BPE;
    const int j   = blockIdx.x % BPE;
    const int tid = threadIdx.x;
    const int wv  = tid >> 5;
    const int ln  = tid & 31;

    // ---- Stage W[e] (128x128) into LDS, K-paired ----
    const float* We = W + (size_t)e * D_IN * D_OUT;
    {
        // 64 kpairs x 64 column-pairs = 4096 work items, 16 per thread
        #pragma unroll
        for (int i = 0; i < 16; ++i) {
            int q  = i * GEMM_BLK + tid;
            int p  = q >> 6;                 // kpair 0..63
            int n2 = (q & 63) << 1;          // column 0,2,..,126
            float2 r0 = *(const float2*)(We + (2 * p)     * D_OUT + n2);
            float2 r1 = *(const float2*)(We + (2 * p + 1) * D_OUT + n2);
            float* d = &w_pair[p * WP_STRIDE + 2 * n2];
            *(float2*)(d)     = make_float2(r0.x, r1.x);
            *(float2*)(d + 2) = make_float2(r0.y, r1.y);
        }
    }
    __syncthreads();

    const int segStart = ws[WS_SEG + e];
    const int Te       = ws[WS_TILES + e];
    const int* perm    = ws + WS_PERM;

    const int half   = (ln < 16) ? 0 : 2;   // K sub-offset for A
    const int kph    = (ln < 16) ? 0 : 1;   // kpair sub-offset for B
    const int lane16 = ln & 15;

    // Per-lane bias for the 8 N-tiles (lanes 16-31 mirror 0-15, matching C layout)
    float bcol[8];
    #pragma unroll
    for (int nt = 0; nt < 8; ++nt) bcol[nt] = bias[e * D_OUT + nt * 16 + lane16];

    float* my_a = a_lds + wv * (TILE_M * A_LDS_STRIDE);
    const int tile_step = BPE * 8;

    for (int tile = j * 8 + wv; tile < Te; tile += tile_step) {
        const int tbase = segStart + tile * TILE_M;

        // Hint the next tile's perm entries toward the caches.
        if (tile + tile_step < Te)
            __builtin_prefetch(&perm[segStart + (tile + tile_step) * TILE_M], 0, 1);

        // ---- Gather A: 16 token rows of x into this wave's LDS slab ----
        #pragma unroll 4
        for (int r = 0; r < TILE_M; ++r) {
            int tok = perm[tbase + r];
            float4 v = make_float4(0.f, 0.f, 0.f, 0.f);
            if (tok >= 0)
                v = *(const float4*)(x + (size_t)tok * D_IN + (ln << 2));
            float* d = &my_a[r * A_LDS_STRIDE + (ln << 2)];
            d[0] = v.x; d[1] = v.y; d[2] = v.z; d[3] = v.w;
        }
        // DS ops from one wave complete in order; fence compiler + counter.
        asm volatile("s_wait_dscnt 0" ::: "memory");

        v8f acc[8];
        #pragma unroll
        for (int nt = 0; nt < 8; ++nt) acc[nt] = (v8f)(0.f);

        // ---- K reduction: 32 chunks of K=4, 8 N-tiles each ----
        #pragma unroll 4
        for (int c = 0; c < D_IN / 4; ++c) {
            const v2f a = *(const v2f*)&my_a[lane16 * A_LDS_STRIDE + c * 4 + half];
            const float* brow = &w_pair[(2 * c + kph) * WP_STRIDE];
            #pragma unroll
            for (int nt = 0; nt < 8; ++nt) {
                const v2f b = *(const v2f*)&brow[(nt * 16 + lane16) * 2];
                acc[nt] = __builtin_amdgcn_wmma_f32_16x16x4_f32(
                    /*neg_a=*/false, a, /*neg_b=*/false, b,
                    /*c_mod=*/(short)0, acc[nt],
                    /*reuse_a=*/false, /*reuse_b=*/false);
            }
        }

        // ---- Store with fused bias. C layout: VGPR i, lanes0-15 -> M=i,
        // lanes16-31 -> M=8+i, N = lane16. ----
        const int mbase = (ln < 16) ? 0 : 8;
        #pragma unroll
        for (int i = 0; i < 8; ++i) {
            int m   = mbase + i;
            int tok = perm[tbase + m];
            if (tok < 0) continue;
            float* orow = out + (size_t)tok * D_OUT;
            #pragma unroll
            for (int nt = 0; nt < 8; ++nt) {
                orow[nt * 16 + lane16] = acc[nt][i] + bcol[nt];
            }
        }
    }
}

extern "C" void kernel_launch(void* const* d_in, const int* in_sizes, int n_in,
                              void* d_out, int out_size, void* d_ws, size_t ws_size,
                              hipStream_t stream) {
    const float*     x     = (const float*)d_in[0];
    const long long* split = (const long long*)d_in[1];
    const float*     W     = (const float*)d_in[2];
    const float*     bias  = (const float*)d_in[3];
    float*           out   = (float*)d_out;
    int*             ws    = (int*)d_ws;

    const int n = in_sizes[1];             // token count (262144)

    moe_init_kernel<<<1, 32, 0, stream>>>(ws);
    moe_hist_kernel<<<(n + 255) / 256, 256, 0, stream>>>(split, ws, n);
    moe_scan_pad_kernel<<<1, 32, 0, stream>>>(ws);
    moe_scatter_kernel<<<(n + 255) / 256, 256, 0, stream>>>(split, ws, n);

    const size_t lds_bytes =
        ((D_IN / 2) * WP_STRIDE + 8 * TILE_M * A_LDS_STRIDE) * sizeof(float); // ~138 KB
    moe_gemm_kernel<<<N_EXPERTS * BPE, GEMM_BLK, lds_bytes, stream>>>(x, W, bias, ws, out);
}